// PolyFourierConverter_25383256719433
// MI455X (gfx1250) — compile-verified
//
#include <hip/hip_runtime.h>
#include <hip/hip_bf16.h>
#include <math.h>

typedef float v2f __attribute__((ext_vector_type(2)));
typedef float v8f __attribute__((ext_vector_type(8)));

#define BATCH 64
#define NTRI  128
#define HW    2048

// Frequency grid: geometric progression 0.1 .. 1.0 over 31 steps, mirrored for X.
__device__ __forceinline__ float freq_x(int h) {
    if (h == 31 || h >= 63) return 0.0f;                // center zero + pad row
    if (h < 31)  return -0.1f * powf(10.0f, (float)(30 - h) * (1.0f / 30.0f));
    return 0.1f * powf(10.0f, (float)(h - 32) * (1.0f / 30.0f));
}
__device__ __forceinline__ float freq_y(int w) {
    if (w == 0) return 0.0f;
    return 0.1f * powf(10.0f, (float)(w - 1) * (1.0f / 30.0f));
}

__global__ __launch_bounds__(256)
void poly_cft_kernel(const float* __restrict__ tris,
                     const int*  __restrict__ lengths,
                     float*      __restrict__ out)
{
    // Coefficient pairs packed for single ds_load_b64; slot NTRI is a zero pad
    // so lanes 16-31 (K=2,3 rows of B, must be 0) read it branchlessly.
    __shared__ float2 s_B1[NTRI + 1];   // (xr-xq, yr-yq)
    __shared__ float2 s_B2[NTRI + 1];   // (xs-xr, ys-yr)
    __shared__ float2 s_B3[NTRI + 1];   // (xq, yq)
    __shared__ float  s_ar[NTRI];

    const int b   = blockIdx.x;
    const int tid = threadIdx.x;

    if (tid < NTRI) {
        const float* t6 = tris + ((size_t)b * NTRI + tid) * 6;
        __builtin_prefetch(t6, 0, 1);                    // global_prefetch_b8
        float xq = t6[0], yq = t6[1];
        float xr = t6[2], yr = t6[3];
        float xs = t6[4], ys = t6[5];
        s_B1[tid] = make_float2(xr - xq, yr - yq);
        s_B2[tid] = make_float2(xs - xr, ys - yr);
        s_B3[tid] = make_float2(xq, yq);
        float det = xq * (yr - ys) + xr * (ys - yq) + xs * (yq - yr);
        s_ar[tid] = fabsf(0.5f * det);
    }
    if (tid == NTRI) {                                   // zero pad slot
        s_B1[NTRI] = make_float2(0.0f, 0.0f);
        s_B2[NTRI] = make_float2(0.0f, 0.0f);
        s_B3[NTRI] = make_float2(0.0f, 0.0f);
    }
    __syncthreads();

    const int lane  = tid & 31;
    const int wave  = tid >> 5;
    const int pbase = blockIdx.y * 128 + wave * 16;      // 16 grid points per wave
    const int nlane = lane & 15;                         // triangle index within chunk
    const int len   = lengths[b];

    // A matrix (16 points x K=4), f32 layout: VGPR0 = {K0 | K2}, VGPR1 = {K1 | K3}.
    // K2/K3 columns zero -> lanes 16-31 hold 0.
    const int pA = pbase + nlane;
    const float fx = freq_x(pA >> 5);
    const float fy = freq_y(pA & 31);
    v2f amat;
    amat[0] = (lane < 16) ? fx : 0.0f;
    amat[1] = (lane < 16) ? fy : 0.0f;

    const float TWO_PI  = 6.2831853071795864769f;
    const float FOURPI2 = 39.478417604357434f;           // 4*pi^2

    float accre[8], accim[8];
    #pragma unroll
    for (int r = 0; r < 8; ++r) { accre[r] = 0.0f; accim[r] = 0.0f; }

    for (int n0 = 0; n0 < NTRI; n0 += 16) {
        if (n0 >= len) break;                            // uniform per wave
        const int t   = n0 + nlane;
        const int idx = (lane < 16) ? t : NTRI;          // high half -> zero slot

        // Branchless B builds: one ds_load_b64 each.
        float2 c1 = s_B1[idx];
        float2 c2 = s_B2[idx];
        float2 c3 = s_B3[idx];
        v2f b1; b1[0] = c1.x; b1[1] = c1.y;
        v2f b2; b2[0] = c2.x; b2[1] = c2.y;
        v2f b3; b3[0] = c3.x; b3[1] = c3.y;

        v8f z = {};
        // D[m,n] = U[m]*c0[n] + V[m]*c1[n]   (rank-2, K=4 zero-padded)
        v8f d1 = __builtin_amdgcn_wmma_f32_16x16x4_f32(false, amat, false, b1, (short)0, z, false, false);
        v8f d2 = __builtin_amdgcn_wmma_f32_16x16x4_f32(false, amat, false, b2, (short)0, z, false, false);
        v8f d3 = __builtin_amdgcn_wmma_f32_16x16x4_f32(false, amat, false, b3, (short)0, z, false, false);

        const float area = s_ar[t];                      // per-lane triangle (N = lane%16)
        const float val  = (t < len) ? 1.0f : 0.0f;

        #pragma unroll
        for (int r = 0; r < 8; ++r) {
            float u  = d1[r], v = d2[r], ps = d3[r];
            float s  = u + v;

            bool uz = (u == 0.0f);
            bool vz = (v == 0.0f);
            bool sz = (s == 0.0f);
            bool zm = uz && vz;                          // zero_mask
            bool dg = sz && !zm;                         // diag_mask
            bool um = uz && !zm;                         // u_mask
            bool vm = vz && !zm;                         // v_mask
            bool nm = !uz && !vz && !sz;                 // normal_mask

            // part1: precompute all arms, then pure cndmask chains (no branches)
            float den_nm = FOURPI2 * u * v * s;
            float den_uu = FOURPI2 * u * u;
            float den_vv = FOURPI2 * v * v;
            float den = 1.0f;
            den = nm ? den_nm : den;
            den = (dg || vm) ? den_uu : den;
            den = um ? den_vv : den;
            float sgn = 0.0f;
            sgn = (nm || vm) ?  1.0f : sgn;
            sgn = (dg || um) ? -1.0f : sgn;
            float p1 = sgn / den;

            // base_u = exp(-2pi i u) = (cu,-su); base_v likewise; base_uv via angle-add
            float su, cu, sv, cv, sp, cp;
            __sincosf(TWO_PI * u,  &su, &cu);
            __sincosf(TWO_PI * v,  &sv, &cv);
            __sincosf(TWO_PI * ps, &sp, &cp);
            float cs = cu * cv - su * sv;                // cos(2pi(u+v))
            float ss = su * cv + cu * sv;                // sin(2pi(u+v))

            // part2 arms (all cheap), then cndmask chains
            float p2re_nm = fmaf(-u, cs, fmaf(s, cu, -v));
            float p2im_nm = fmaf( u, ss, -s * su);
            float p2re_dg = cu - 1.0f;
            float p2im_dg = fmaf(TWO_PI, u, -su);
            float p2re_um = cv - 1.0f;
            float p2im_um = fmaf(TWO_PI, v, -sv);
            float tu      = TWO_PI * u;
            float p2re_vm = fmaf(tu, su, cu) - 1.0f;
            float p2im_vm = fmaf(tu, cu, -su);

            float p2re = 0.0f, p2im = 0.0f;
            p2re = nm ? p2re_nm : p2re;   p2im = nm ? p2im_nm : p2im;
            p2re = dg ? p2re_dg : p2re;   p2im = dg ? p2im_dg : p2im;
            p2re = um ? p2re_um : p2re;   p2im = um ? p2im_um : p2im;
            p2re = vm ? p2re_vm : p2re;   p2im = vm ? p2im_vm : p2im;

            // FT = 2*area*p1 * part2 * exp(-2pi i ps);  zero-case: FT = area
            float k   = 2.0f * area * p1 * val;
            float fre = k * fmaf(p2re, cp,  p2im * sp);
            float fim = k * fmaf(p2im, cp, -p2re * sp);
            fre = zm ? area * val : fre;
            fim = zm ? 0.0f       : fim;

            accre[r] += fre;
            accim[r] += fim;
        }
    }

    // Reduce over triangles (N = lane%16): sum within each 16-lane half.
    #pragma unroll
    for (int r = 0; r < 8; ++r) {
        float re = accre[r], im = accim[r];
        re += __shfl_xor(re, 8, 32);  im += __shfl_xor(im, 8, 32);
        re += __shfl_xor(re, 4, 32);  im += __shfl_xor(im, 4, 32);
        re += __shfl_xor(re, 2, 32);  im += __shfl_xor(im, 2, 32);
        re += __shfl_xor(re, 1, 32);  im += __shfl_xor(im, 1, 32);
        accre[r] = re;  accim[r] = im;
    }

    // Lane 0 of each half writes M = r (low half) / r+8 (high half).
    if (nlane == 0) {
        const int half = lane >> 4;
        #pragma unroll
        for (int r = 0; r < 8; ++r) {
            int   p  = pbase + r + 8 * half;
            int   h  = p >> 5;
            float re = accre[r], im = accim[r];
            if (h == 63) { re = 0.0f; im = 0.0f; }       // pad-row mask
            float m2  = re * re + im * im;
            float mag = 0.0f, phs = 0.0f;
            if (m2 != 0.0f) {
                mag = log1pf(sqrtf(m2));
                phs = atan2f(im, re);
            }
            out[(size_t)b * HW + p]                      = mag;
            out[(size_t)BATCH * HW + (size_t)b * HW + p] = phs;
        }
    }
}

extern "C" void kernel_launch(void* const* d_in, const int* in_sizes, int n_in,
                              void* d_out, int out_size, void* d_ws, size_t ws_size,
                              hipStream_t stream) {
    (void)in_sizes; (void)n_in; (void)d_ws; (void)ws_size; (void)out_size;
    const float* tris    = (const float*)d_in[0];   // [64,128,3,2] f32
    const int*   lengths = (const int*)  d_in[1];   // [64] i32
    float*       out     = (float*)d_out;           // mag[64*2048] ++ phase[64*2048]

    dim3 grid(BATCH, 16);   // 64 batches x (16 blocks * 8 waves * 16 points = 2048 points)
    poly_cft_kernel<<<grid, 256, 0, stream>>>(tris, lengths, out);
}